// Subgraph_sampler_46033459479300
// MI455X (gfx1250) — compile-verified
//
#include <hip/hip_runtime.h>
#include <math.h>

// Problem constants (fixed by the reference)
#define NNODES 102400
#define NGRAPH 512
#define HIDC   256
#define TWOH   512
#define NPG    200
#define DEG    32
#define NEDGE  (NNODES * DEG)

typedef float v2f __attribute__((ext_vector_type(2)));
typedef float v8f __attribute__((ext_vector_type(8)));

// ---------------------------------------------------------------------------
// Kernel 1: fused  x = h + g[batch];  z = x@W1 + b1;  LayerNorm;  ReLU;
//           logits = z@W2 + b2.   GEMM via exact-fp32 V_WMMA_F32_16X16X4_F32.
// Block = 256 threads (8 wave32s), 16 nodes per block.
// ---------------------------------------------------------------------------
__global__ __launch_bounds__(256) void mlp_logits_kernel(
    const float* __restrict__ h, const float* __restrict__ g,
    const int* __restrict__ batch, const float* __restrict__ W1,
    const float* __restrict__ b1, const float* __restrict__ ln_g,
    const float* __restrict__ ln_b, const float* __restrict__ W2,
    const float* __restrict__ b2, float* __restrict__ logits)
{
    __shared__ float xs[16][HIDC];   // 16 KB: input tile
    __shared__ float zs[16][TWOH];   // 32 KB: hidden tile

    const int t    = threadIdx.x;
    const int wave = t >> 5;
    const int lane = t & 31;
    const int half = lane >> 4;      // 0: lanes 0-15, 1: lanes 16-31
    const int l16  = lane & 15;
    const int row0 = blockIdx.x * 16;

    // Stage x = h + g[batch] (each thread loads one column of each row).
    for (int r = 0; r < 16; ++r) {
        const int node = row0 + r;
        const int b = batch[node];
        xs[r][t] = h[(size_t)node * HIDC + t] + g[(size_t)b * HIDC + t];
    }
    __syncthreads();

    // Each wave computes 4 of the 32 16x16 column tiles of z = x@W1.
    for (int j = 0; j < 4; ++j) {
        const int n0 = (wave + 8 * j) * 16;
        v8f c = {};
        #pragma unroll 4
        for (int k0 = 0; k0 < HIDC; k0 += 4) {
            // A (16x4 f32): lanes 0-15 hold K=k0,k0+1; lanes 16-31 hold k0+2,k0+3
            v2f a, b;
            a.x = xs[l16][k0 + 2 * half];
            a.y = xs[l16][k0 + 2 * half + 1];
            // B (4x16 f32): mirrored layout, rows of W1 (row-major [HID][2H])
            b.x = W1[(size_t)(k0 + 2 * half)     * TWOH + n0 + l16];
            b.y = W1[(size_t)(k0 + 2 * half + 1) * TWOH + n0 + l16];
            c = __builtin_amdgcn_wmma_f32_16x16x4_f32(
                    false, a, false, b, (short)0, c, false, false);
        }
        // C/D layout: VGPR r, lanes 0-15 -> M=r, lanes 16-31 -> M=r+8
        #pragma unroll
        for (int r = 0; r < 8; ++r) {
            const int m = r + 8 * half;
            zs[m][n0 + l16] = c[r] + b1[n0 + l16];
        }
    }
    __syncthreads();

    // LayerNorm + ReLU + GEMV with W2. Wave w handles rows 2w, 2w+1.
    for (int rr = 0; rr < 2; ++rr) {
        const int row = wave * 2 + rr;
        float s = 0.f, sq = 0.f;
        #pragma unroll
        for (int jj = 0; jj < 16; ++jj) {
            const float v = zs[row][lane + 32 * jj];   // stride-1 across lanes
            s += v; sq += v * v;
        }
        for (int off = 16; off > 0; off >>= 1) {
            s  += __shfl_xor(s,  off, 32);
            sq += __shfl_xor(sq, off, 32);
        }
        const float mean = s  * (1.f / TWOH);
        const float var  = sq * (1.f / TWOH) - mean * mean;   // biased var
        const float rstd = rsqrtf(var + 1e-5f);
        float acc = 0.f;
        #pragma unroll
        for (int jj = 0; jj < 16; ++jj) {
            const int cidx = lane + 32 * jj;
            float v = (zs[row][cidx] - mean) * rstd * ln_g[cidx] + ln_b[cidx];
            v = v > 0.f ? v : 0.f;                            // ReLU
            acc += v * W2[cidx];
        }
        for (int off = 16; off > 0; off >>= 1) acc += __shfl_xor(acc, off, 32);
        if (lane == 0) logits[row0 + row] = acc + b2[0];
    }
}

// ---------------------------------------------------------------------------
// Kernel 2: per-graph softmax + top-1 seed (first max wins). 1 block / graph.
// ---------------------------------------------------------------------------
__global__ __launch_bounds__(256) void segment_softmax_kernel(
    const float* __restrict__ logits, float* __restrict__ node_prob,
    int* __restrict__ nm0)
{
    __shared__ float redf[256];
    __shared__ int   redi[256];
    __shared__ float s_max, s_sum, s_pmax;

    const int t    = threadIdx.x;
    const int node = blockIdx.x * NPG + t;
    const bool valid = t < NPG;

    const float lg = valid ? logits[node] : -3.0e38f;
    redf[t] = lg; __syncthreads();
    for (int o = 128; o > 0; o >>= 1) { if (t < o) redf[t] = fmaxf(redf[t], redf[t + o]); __syncthreads(); }
    if (t == 0) s_max = redf[0];
    __syncthreads();

    const float e = valid ? expf(lg - s_max) : 0.f;
    redf[t] = e; __syncthreads();
    for (int o = 128; o > 0; o >>= 1) { if (t < o) redf[t] += redf[t + o]; __syncthreads(); }
    if (t == 0) s_sum = redf[0];
    __syncthreads();

    const float prob = valid ? e / s_sum : -1.f;
    if (valid) node_prob[node] = prob;
    redf[t] = prob; __syncthreads();
    for (int o = 128; o > 0; o >>= 1) { if (t < o) redf[t] = fmaxf(redf[t], redf[t + o]); __syncthreads(); }
    if (t == 0) s_pmax = redf[0];
    __syncthreads();

    redi[t] = (valid && prob >= s_pmax) ? node : 0x7fffffff;
    __syncthreads();
    for (int o = 128; o > 0; o >>= 1) { if (t < o) redi[t] = min(redi[t], redi[t + o]); __syncthreads(); }
    if (t == 0) nm0[redi[0]] = 1;     // seed the BFS mask
}

// ---------------------------------------------------------------------------
// BFS helpers
// ---------------------------------------------------------------------------
__global__ void zero_init_kernel(int* __restrict__ nm0, float* __restrict__ nmask)
{
    const int i = blockIdx.x * blockDim.x + threadIdx.x;
    if (i < NNODES) { nm0[i] = 0; nmask[i] = 0.f; }
}

__global__ void copy_mask_kernel(const int* __restrict__ s, int* __restrict__ d)
{
    const int i = blockIdx.x * blockDim.x + threadIdx.x;
    if (i < NNODES) d[i] = s[i];
}

// nm_next (pre-copied from nm_prev) |= scatter(src <- nm_prev[dst])
__global__ void hop_kernel(const int* __restrict__ src, const int* __restrict__ dst,
                           const int* __restrict__ nm_prev, int* __restrict__ nm_next)
{
    const int e = blockIdx.x * blockDim.x + threadIdx.x;
    if (e < NEDGE && nm_prev[dst[e]]) nm_next[src[e]] = 1;  // benign same-value race
}

__global__ void finalize_kernel(const int* __restrict__ src, const int* __restrict__ dst,
                                const int* __restrict__ nm, const float* __restrict__ logits,
                                float* __restrict__ ew, float* __restrict__ em,
                                float* __restrict__ nmask)
{
    const int e = blockIdx.x * blockDim.x + threadIdx.x;
    if (e >= NEDGE) return;
    const int s = src[e], d = dst[e];
    ew[e] = logits[s] + logits[d];
    const bool keep = (nm[s] != 0) && (nm[d] != 0);
    em[e] = keep ? 1.f : 0.f;
    if (keep) { nmask[s] = 1.f; nmask[d] = 1.f; }           // benign same-value race
}

// ---------------------------------------------------------------------------
extern "C" void kernel_launch(void* const* d_in, const int* in_sizes, int n_in,
                              void* d_out, int out_size, void* d_ws, size_t ws_size,
                              hipStream_t stream)
{
    const float* h     = (const float*)d_in[0];
    const float* g     = (const float*)d_in[1];
    const int*   batch = (const int*)  d_in[2];
    const int*   ei    = (const int*)  d_in[3];   // [2, E]
    const float* W1    = (const float*)d_in[4];
    const float* b1    = (const float*)d_in[5];
    const float* ln_g  = (const float*)d_in[6];
    const float* ln_b  = (const float*)d_in[7];
    const float* W2    = (const float*)d_in[8];
    const float* b2    = (const float*)d_in[9];
    const int* src = ei;
    const int* dst = ei + NEDGE;

    // Outputs concatenated: node_prob[N], edge_weight[E], edge_mask[E], node_mask[N]
    float* out       = (float*)d_out;
    float* node_prob = out;
    float* ew        = out + (size_t)NNODES;
    float* em        = out + (size_t)NNODES + (size_t)NEDGE;
    float* nmask     = out + (size_t)NNODES + 2 * (size_t)NEDGE;

    // Workspace: logits[N] f32, nm0[N] i32, nm1[N] i32  (~1.2 MB)
    float* logits = (float*)d_ws;
    int*   nm0    = (int*)((char*)d_ws + (size_t)NNODES * 4);
    int*   nm1    = (int*)((char*)d_ws + (size_t)NNODES * 8);

    const int nb_node = (NNODES + 255) / 256;
    const int nb_edge = (NEDGE + 255) / 256;

    mlp_logits_kernel<<<NNODES / 16, 256, 0, stream>>>(
        h, g, batch, W1, b1, ln_g, ln_b, W2, b2, logits);
    zero_init_kernel<<<nb_node, 256, 0, stream>>>(nm0, nmask);
    segment_softmax_kernel<<<NGRAPH, 256, 0, stream>>>(logits, node_prob, nm0);

    // hop 1: nm1 = max(nm0, scatter(src <- nm0[dst]))
    copy_mask_kernel<<<nb_node, 256, 0, stream>>>(nm0, nm1);
    hop_kernel<<<nb_edge, 256, 0, stream>>>(src, dst, nm0, nm1);
    // hop 2: nm0 = max(nm1, scatter(src <- nm1[dst]))
    copy_mask_kernel<<<nb_node, 256, 0, stream>>>(nm1, nm0);
    hop_kernel<<<nb_edge, 256, 0, stream>>>(src, dst, nm1, nm0);

    finalize_kernel<<<nb_edge, 256, 0, stream>>>(src, dst, nm0, logits, ew, em, nmask);
}